// CGCNN_48180943127383
// MI455X (gfx1250) — compile-verified
//
#include <hip/hip_runtime.h>
#include <hip/hip_bf16.h>
#include <math.h>

#define GRAPHS 64

typedef __attribute__((ext_vector_type(2))) float v2f;
typedef __attribute__((ext_vector_type(8))) float v8f;

__device__ __forceinline__ void atomicMaxF(float* addr, float v) {
  // classic bit-pattern trick: works for any sign mix when init to -inf
  if (v >= 0.0f) atomicMax((int*)addr, __float_as_int(v));
  else           atomicMin((unsigned int*)addr, __float_as_uint(v));
}

__global__ void k_copy(const float* __restrict__ src, float* __restrict__ dst, int n) {
  int i = blockIdx.x * blockDim.x + threadIdx.x;
  if (i < n) dst[i] = src[i];
}

// pools layout: mx1 sm1 mx2 sm2 mx3 sm3 (192 each) | counts(64) | invc(64)
__global__ void k_init_small(float* p) {
  int i = blockIdx.x * blockDim.x + threadIdx.x;
  if (i < 6 * GRAPHS * 3) {
    int chunk = i / (GRAPHS * 3);
    p[i] = (chunk & 1) ? 0.0f : -INFINITY;  // even chunks are max buffers
  } else if (i < 6 * GRAPHS * 3 + GRAPHS) {
    p[i] = 0.0f;                            // counts
  }
}

__global__ __launch_bounds__(256) void k_counts(const int* __restrict__ batch,
                                                float* __restrict__ counts, int n) {
  __shared__ float lc[GRAPHS];
  if (threadIdx.x < GRAPHS) lc[threadIdx.x] = 0.0f;
  __syncthreads();
  int i = blockIdx.x * blockDim.x + threadIdx.x;
  if (i < n) atomicAdd(&lc[batch[i]], 1.0f);
  __syncthreads();
  if (threadIdx.x < GRAPHS && lc[threadIdx.x] != 0.0f)
    atomicAdd(&counts[threadIdx.x], lc[threadIdx.x]);
}

__global__ void k_inv(const float* __restrict__ counts, float* __restrict__ invc) {
  int g = threadIdx.x;
  if (g < GRAPHS) invc[g] = 1.0f / fmaxf(counts[g], 1.0f);
}

// CGConv edge kernel: hout must be pre-initialized to xin (residual); atomics accumulate.
__global__ __launch_bounds__(256) void k_conv(
    const float* __restrict__ xin, float* __restrict__ hout,
    const int* __restrict__ ei, const float* __restrict__ ea,
    const float* __restrict__ Wf, const float* __restrict__ bf,
    const float* __restrict__ Ws, const float* __restrict__ bs, int ecount) {
  float wf[10][3], wsm[10][3], bfv[3], bsv[3];
#pragma unroll
  for (int k = 0; k < 10; ++k)
#pragma unroll
    for (int c = 0; c < 3; ++c) { wf[k][c] = Wf[k * 3 + c]; wsm[k][c] = Ws[k * 3 + c]; }
#pragma unroll
  for (int c = 0; c < 3; ++c) { bfv[c] = bf[c]; bsv[c] = bs[c]; }

  int e = blockIdx.x * blockDim.x + threadIdx.x;
  if (e >= ecount) return;
  int s = ei[e];
  int d = ei[ecount + e];
  float z[10];
  z[0] = xin[d * 3 + 0]; z[1] = xin[d * 3 + 1]; z[2] = xin[d * 3 + 2];
  z[3] = xin[s * 3 + 0]; z[4] = xin[s * 3 + 1]; z[5] = xin[s * 3 + 2];
  const float4 ev = *(const float4*)(ea + (size_t)e * 4);
  z[6] = ev.x; z[7] = ev.y; z[8] = ev.z; z[9] = ev.w;
#pragma unroll
  for (int c = 0; c < 3; ++c) {
    float a = bfv[c], b = bsv[c];
#pragma unroll
    for (int k = 0; k < 10; ++k) { a = fmaf(z[k], wf[k][c], a); b = fmaf(z[k], wsm[k][c], b); }
    float sig = 1.0f / (1.0f + __expf(-a));
    float sp  = fmaxf(b, 0.0f) + log1pf(__expf(-fabsf(b)));   // stable softplus
    atomicAdd(&hout[d * 3 + c], sig * sp);
  }
}

// LDS-staged segment max/sum pooling (batch sorted -> few graphs per block)
__global__ __launch_bounds__(256) void k_pool(
    const float* __restrict__ h, const int* __restrict__ batch,
    float* __restrict__ mx, float* __restrict__ sm, int n) {
  __shared__ float lmx[GRAPHS * 3];
  __shared__ float lsm[GRAPHS * 3];
  if (threadIdx.x < GRAPHS * 3) { lmx[threadIdx.x] = -INFINITY; lsm[threadIdx.x] = 0.0f; }
  __syncthreads();
  int i = blockIdx.x * blockDim.x + threadIdx.x;
  if (i < n) {
    int b = batch[i];
#pragma unroll
    for (int c = 0; c < 3; ++c) {
      float v = h[i * 3 + c];
      atomicMaxF(&lmx[b * 3 + c], v);
      atomicAdd(&lsm[b * 3 + c], v);
    }
  }
  __syncthreads();
  if (threadIdx.x < GRAPHS * 3) {
    float m = lmx[threadIdx.x];
    if (m > -INFINITY) atomicMaxF(&mx[threadIdx.x], m);
    float sv = lsm[threadIdx.x];
    if (sv != 0.0f) atomicAdd(&sm[threadIdx.x], sv);
  }
}

// Head: x4[64,18] -> x4@linW+b -> [x4l|x4][64,82] -> @clsW+b -> out[64,10]
// Single wave32, fp32 WMMA 16x16x4 with accumulate chaining.
__global__ __launch_bounds__(32) void k_head(
    const float* __restrict__ pools, const float* __restrict__ invc,
    const float* __restrict__ linW, const float* __restrict__ linB,
    const float* __restrict__ clsW, const float* __restrict__ clsB,
    float* __restrict__ out) {
  __shared__ float X4[64 * 20];  // K padded 18 -> 20
  __shared__ float H[64 * 84];   // K padded 82 -> 84
  const int lane = threadIdx.x;
  // build x4: col c -> stage c/6, (max if c%6<3 else mean)
  for (int i = lane; i < 64 * 20; i += 32) {
    int g = i / 20, c = i % 20;
    float v = 0.0f;
    if (c < 18) {
      int stage = c / 6, w = c % 6;
      const float* mxp = pools + stage * 384;
      const float* smp = mxp + 192;
      v = (w < 3) ? mxp[g * 3 + w] : smp[g * 3 + (w - 3)] * invc[g];
    }
    X4[i] = v;
  }
  __syncthreads();
  const int mhalf = lane >> 4;  // WMMA f32 A layout: lanes 16-31 hold K+2
  const int mn    = lane & 15;
  // GEMM1: [64x18] @ [18x64] -> H[:,0:64]
  for (int mt = 0; mt < 4; ++mt) {
    for (int nt = 0; nt < 4; ++nt) {
      v8f acc = {};
      for (int kt = 0; kt < 5; ++kt) {
        int k0 = kt * 4 + mhalf * 2;
        v2f a, b;
        a.x = X4[(mt * 16 + mn) * 20 + k0];
        a.y = X4[(mt * 16 + mn) * 20 + k0 + 1];
        int ka = (k0 < 17) ? k0 : 17;
        int kb = (k0 + 1 < 17) ? (k0 + 1) : 17;
        b.x = (k0     < 18) ? linW[ka * 64 + nt * 16 + mn] : 0.0f;
        b.y = (k0 + 1 < 18) ? linW[kb * 64 + nt * 16 + mn] : 0.0f;
        acc = __builtin_amdgcn_wmma_f32_16x16x4_f32(false, a, false, b,
                                                    (short)0, acc, false, false);
      }
      float bb = linB[nt * 16 + mn];
#pragma unroll
      for (int r = 0; r < 8; ++r)
        H[(mt * 16 + mhalf * 8 + r) * 84 + nt * 16 + mn] = acc[r] + bb;
    }
  }
  // H[:,64:84] = [x4 | 0pad]
  for (int i = lane; i < 64 * 20; i += 32) {
    int g = i / 20, c = i % 20;
    H[g * 84 + 64 + c] = X4[i];
  }
  __syncthreads();
  // GEMM2: [64x84] @ [84x16(pad of 82x10)] -> out[64x10]
  for (int mt = 0; mt < 4; ++mt) {
    v8f acc = {};
    for (int kt = 0; kt < 21; ++kt) {
      int k0 = kt * 4 + mhalf * 2;
      v2f a, b;
      a.x = H[(mt * 16 + mn) * 84 + k0];
      a.y = H[(mt * 16 + mn) * 84 + k0 + 1];
      int ka = (k0 < 81) ? k0 : 81;
      int kb = (k0 + 1 < 81) ? (k0 + 1) : 81;
      int nc = (mn < 9) ? mn : 9;
      b.x = ((k0     < 82) && (mn < 10)) ? clsW[ka * 10 + nc] : 0.0f;
      b.y = ((k0 + 1 < 82) && (mn < 10)) ? clsW[kb * 10 + nc] : 0.0f;
      acc = __builtin_amdgcn_wmma_f32_16x16x4_f32(false, a, false, b,
                                                  (short)0, acc, false, false);
    }
    if (mn < 10) {
      float bb = clsB[mn];
#pragma unroll
      for (int r = 0; r < 8; ++r)
        out[(mt * 16 + mhalf * 8 + r) * 10 + mn] = acc[r] + bb;
    }
  }
}

extern "C" void kernel_launch(void* const* d_in, const int* in_sizes, int n_in,
                              void* d_out, int out_size, void* d_ws, size_t ws_size,
                              hipStream_t stream) {
  const float* x     = (const float*)d_in[0];
  const int*   ei    = (const int*)d_in[1];
  const float* ea    = (const float*)d_in[2];
  const int*   batch = (const int*)d_in[3];
  const float* Wf1 = (const float*)d_in[4];
  const float* bf1 = (const float*)d_in[5];
  const float* Ws1 = (const float*)d_in[6];
  const float* bs1 = (const float*)d_in[7];
  const float* Wf2 = (const float*)d_in[8];
  const float* bf2 = (const float*)d_in[9];
  const float* Ws2 = (const float*)d_in[10];
  const float* bs2 = (const float*)d_in[11];
  const float* linW = (const float*)d_in[12];
  const float* linB = (const float*)d_in[13];
  const float* clsW = (const float*)d_in[14];
  const float* clsB = (const float*)d_in[15];
  float* out = (float*)d_out;

  const int nodeN = in_sizes[0] / 3;
  const int edgeE = in_sizes[2] / 4;

  float* ws = (float*)d_ws;
  float* hA = ws;
  float* hB = ws + (size_t)nodeN * 3;
  float* pools = hB + (size_t)nodeN * 3;       // 6*192 + 64 + 64
  float* mx1 = pools,        *sm1 = pools + 192;
  float* mx2 = pools + 384,  *sm2 = pools + 576;
  float* mx3 = pools + 768,  *sm3 = pools + 960;
  float* counts = pools + 1152;
  float* invc   = counts + 64;

  int nb3 = (nodeN * 3 + 255) / 256;
  int nbN = (nodeN + 255) / 256;
  int nbE = (edgeE + 255) / 256;

  k_init_small<<<2, 640, 0, stream>>>(pools);
  k_counts<<<nbN, 256, 0, stream>>>(batch, counts, nodeN);
  k_inv<<<1, 64, 0, stream>>>(counts, invc);

  // conv1 (in=x, out=hA)
  k_copy<<<nb3, 256, 0, stream>>>(x, hA, nodeN * 3);
  k_conv<<<nbE, 256, 0, stream>>>(x, hA, ei, ea, Wf1, bf1, Ws1, bs1, edgeE);
  k_pool<<<nbN, 256, 0, stream>>>(hA, batch, mx1, sm1, nodeN);

  // conv2 (in=hA, out=hB)
  k_copy<<<nb3, 256, 0, stream>>>(hA, hB, nodeN * 3);
  k_conv<<<nbE, 256, 0, stream>>>(hA, hB, ei, ea, Wf2, bf2, Ws2, bs2, edgeE);
  k_pool<<<nbN, 256, 0, stream>>>(hB, batch, mx2, sm2, nodeN);

  // conv3 = conv2 reused (in=hB, out=hA)
  k_copy<<<nb3, 256, 0, stream>>>(hB, hA, nodeN * 3);
  k_conv<<<nbE, 256, 0, stream>>>(hB, hA, ei, ea, Wf2, bf2, Ws2, bs2, edgeE);
  k_pool<<<nbN, 256, 0, stream>>>(hA, batch, mx3, sm3, nodeN);

  k_head<<<1, 32, 0, stream>>>(pools, invc, linW, linB, clsW, clsB, out);
}